// KnnGraph_32650341384590
// MI455X (gfx1250) — compile-verified
//
#include <hip/hip_runtime.h>

typedef __attribute__((ext_vector_type(2))) float v2f;
typedef __attribute__((ext_vector_type(8))) float v8f;

#define BATCH  2
#define NPTS   8192
#define NDIM   64
#define KNN    16
#define NTILES (NPTS / 16)

// Prep: transpose x[b][c][n] -> point-major WMMA-swizzled xtsw[b][n][64],
// and nhn[b][n] = -0.5 * ||x_n||^2.
// Swizzle: xtsw[n][h*32 + 2*c + j] = x_n[4*c + 2*h + j]  (h = lane>>4, c = K-chunk, j = pair)
// so each lane's 32 needed floats are contiguous -> b128 loads.
__global__ void __launch_bounds__(256) knn_prep(const float* __restrict__ x,
                                                float* __restrict__ xtsw,
                                                float* __restrict__ nhn) {
  int gid = blockIdx.x * blockDim.x + threadIdx.x;
  if (gid >= BATCH * NPTS) return;
  int b = gid / NPTS;
  int n = gid - b * NPTS;
  const float* xp = x + (size_t)b * NDIM * NPTS + n;   // stride NPTS between channels
  float v[NDIM];
  float s = 0.f;
#pragma unroll
  for (int c = 0; c < NDIM; ++c) {
    float t = xp[(size_t)c * NPTS];   // coalesced across threads (n contiguous)
    v[c] = t;
    s += t * t;
  }
  float* o = xtsw + (size_t)gid * NDIM;
#pragma unroll
  for (int h = 0; h < 2; ++h)
#pragma unroll
    for (int c = 0; c < 16; ++c)
#pragma unroll
      for (int j = 0; j < 2; ++j)
        o[h * 32 + 2 * c + j] = v[4 * c + 2 * h + j];
  nhn[gid] = -0.5f * s;
}

// Load one candidate tile's A fragments (16 x v2f) and accumulator init
// (-0.5*||c||^2 for this lane's 8 candidates, M = v + 8h).
static __device__ __forceinline__ void load_tile(const float* __restrict__ xb,
                                                 const float* __restrict__ nb,
                                                 int l16, int h, int t,
                                                 v2f (&A)[16], float (&nrm)[8]) {
  const v2f* ap = (const v2f*)(xb + (size_t)(t * 16 + l16) * NDIM + h * 32);
#pragma unroll
  for (int c = 0; c < 16; ++c) A[c] = ap[c];
  const float* np = nb + t * 16 + h * 8;
#pragma unroll
  for (int v = 0; v < 8; ++v) nrm[v] = np[v];
}

// 16 chained f32 WMMAs (K=64), then per-lane top-16 update (largest D wins).
static __device__ __forceinline__ void proc_tile(const v2f (&A)[16], const v2f (&bq)[16],
                                                 const float (&nrm)[8], int t, int h,
                                                 float (&kd)[KNN], int (&ki)[KNN],
                                                 float& worst, int& wslot) {
  v8f acc;
#pragma unroll
  for (int v = 0; v < 8; ++v) acc[v] = nrm[v];
#pragma unroll
  for (int c = 0; c < 16; ++c)
    acc = __builtin_amdgcn_wmma_f32_16x16x4_f32(
        /*neg_a=*/false, A[c], /*neg_b=*/false, bq[c],
        /*c_mod=*/(short)0, acc, /*reuse_a=*/false, /*reuse_b=*/false);

  const int cbase = t * 16 + h * 8;
#pragma unroll
  for (int v = 0; v < 8; ++v) {
    float d = acc[v];
    if (d > worst) {
      const int idx = cbase + v;
#pragma unroll
      for (int j = 0; j < KNN; ++j) {
        kd[j] = (j == wslot) ? d : kd[j];
        ki[j] = (j == wslot) ? idx : ki[j];
      }
      worst = kd[0]; wslot = 0;
#pragma unroll
      for (int j = 1; j < KNN; ++j)
        if (kd[j] < worst) { worst = kd[j]; wslot = j; }
    }
  }
}

// One wave32 per block; block = (batch b, query tile of 16 rows).
// WMMA: M = candidate, N = query. acc starts at -0.5||c||^2 so
// D[c][q] = c.q - 0.5||c||^2 ; nearest neighbor == LARGEST D.
// Candidate tiles are double-buffered: buffer X's loads for tile t+2 issue
// right after buffer X's WMMA chain for tile t, giving a full chain of
// 16 WMMAs + selection to hide the L2 round trip.
__global__ void __launch_bounds__(32) knn_main(const float* __restrict__ xtsw,
                                               const float* __restrict__ nhn,
                                               int* __restrict__ out) {
  __shared__ float sd[32][KNN];
  __shared__ int   si[32][KNN];

  const int lane = threadIdx.x;        // 0..31
  const int h    = lane >> 4;
  const int l16  = lane & 15;

  const int bid = blockIdx.x;          // 0 .. BATCH*NTILES-1
  const int b   = bid / NTILES;
  const int qt  = bid - b * NTILES;
  const int q   = qt * 16 + l16;       // this lane's query row

  const float* xb = xtsw + (size_t)b * NPTS * NDIM;
  const float* nb = nhn + (size_t)b * NPTS;

  // Load B operand (query fragments) once: 16 chunks x v2f per lane.
  v2f bq[16];
  {
    const v2f* qp = (const v2f*)(xb + (size_t)q * NDIM + h * 32);
#pragma unroll
    for (int c = 0; c < 16; ++c) bq[c] = qp[c];
  }

  float kd[KNN];
  int   ki[KNN];
#pragma unroll
  for (int j = 0; j < KNN; ++j) { kd[j] = -__builtin_inff(); ki[j] = 0; }
  float worst = -__builtin_inff();
  int   wslot = 0;

  // Software pipeline over candidate tiles (NTILES is even).
  v2f   A0[16], A1[16];
  float n0[8],  n1[8];
  load_tile(xb, nb, l16, h, 0, A0, n0);
  load_tile(xb, nb, l16, h, 1, A1, n1);

  for (int t = 0; t < NTILES - 2; t += 2) {
    proc_tile(A0, bq, n0, t,     h, kd, ki, worst, wslot);
    load_tile(xb, nb, l16, h, t + 2, A0, n0);
    proc_tile(A1, bq, n1, t + 1, h, kd, ki, worst, wslot);
    load_tile(xb, nb, l16, h, t + 3, A1, n1);
  }
  proc_tile(A0, bq, n0, NTILES - 2, h, kd, ki, worst, wslot);
  proc_tile(A1, bq, n1, NTILES - 1, h, kd, ki, worst, wslot);

  // Merge the two candidate halves (lane l16 and l16+16) per query via LDS.
#pragma unroll
  for (int j = 0; j < KNN; ++j) { sd[lane][j] = kd[j]; si[lane][j] = ki[j]; }
  __syncthreads();

  if (lane < 16) {
    float fd[2 * KNN];
    int   fi[2 * KNN];
#pragma unroll
    for (int j = 0; j < KNN; ++j) {
      fd[j]       = sd[l16][j];       fi[j]       = si[l16][j];
      fd[KNN + j] = sd[l16 + 16][j];  fi[KNN + j] = si[l16 + 16][j];
    }
    // Output: (2, B, N, k) int32. Part 0 = nn_idx sorted by ascending distance
    // (= descending D, ties -> lower index), part 1 = center index q.
    int* nn  = out + ((size_t)(0 * BATCH + b) * NPTS + q) * KNN;
    int* ctr = out + ((size_t)(1 * BATCH + b) * NPTS + q) * KNN;
#pragma unroll
    for (int sel = 0; sel < KNN; ++sel) {
      float bv = fd[0]; int bi = fi[0]; int bp = 0;
#pragma unroll
      for (int j = 1; j < 2 * KNN; ++j) {
        bool better = (fd[j] > bv) || ((fd[j] == bv) && (fi[j] < bi));
        if (better) { bv = fd[j]; bi = fi[j]; bp = j; }
      }
      nn[sel]  = bi;
      ctr[sel] = q;
#pragma unroll
      for (int j = 0; j < 2 * KNN; ++j)
        if (j == bp) fd[j] = -__builtin_inff();
    }
  }
}

extern "C" void kernel_launch(void* const* d_in, const int* in_sizes, int n_in,
                              void* d_out, int out_size, void* d_ws, size_t ws_size,
                              hipStream_t stream) {
  const float* x = (const float*)d_in[0];          // (2, 64, 8192, 1) f32
  float* xtsw = (float*)d_ws;                                                        // 4 MB
  float* nhn  = (float*)((char*)d_ws + (size_t)BATCH * NPTS * NDIM * sizeof(float)); // 64 KB
  int* out = (int*)d_out;                          // (2, 2, 8192, 16) int32

  knn_prep<<<(BATCH * NPTS + 255) / 256, 256, 0, stream>>>(x, xtsw, nhn);
  knn_main<<<BATCH * NTILES, 32, 0, stream>>>(xtsw, nhn, out);
}